// OutputLayer_1821066133818
// MI455X (gfx1250) — compile-verified
//
#include <hip/hip_runtime.h>

#define B_TOTAL 32768
#define V_N     1024
#define EDGE_N  6144
#define TILE_B  16
#define THREADS 256
#define ACC_STRIDE 1024   // acc[r][v], 16*1024*4 = 64KB LDS

typedef float v2f __attribute__((ext_vector_type(2)));
typedef float v8f __attribute__((ext_vector_type(8)));

// Recover edge -> v-node index from the one-hot mask (fully coalesced, one
// writer per edge row since each row is exactly one-hot).
__global__ __launch_bounds__(THREADS)
void build_edge_index(const float* __restrict__ mask, int* __restrict__ idx) {
    const int i = blockIdx.x * THREADS + threadIdx.x;   // over EDGE_N * V_N (exact multiple)
    if (mask[i] != 0.0f) idx[i >> 10] = i & (V_N - 1);
}

// One block = 16 batch rows x all 1024 v-nodes.
// Phase 1: zero LDS accumulator.
// Phase 2: stream from_prev coalesced along edges, scatter via ds_add_f32.
// Phase 3: merge acc + from_input per 16x16 tile with chained
//          V_WMMA_F32_16X16X4_F32 (A = acc slice, B = identity slice, C = input).
__global__ __launch_bounds__(THREADS)
void ldpc_vnode_update(const float* __restrict__ from_input,
                       const float* __restrict__ from_prev,
                       const int*   __restrict__ idx,
                       float*       __restrict__ out) {
    extern __shared__ float acc[];                 // [TILE_B][ACC_STRIDE]
    const int tid = threadIdx.x;
    const int b0  = blockIdx.x * TILE_B;

    // ---- Phase 1: zero accumulator -------------------------------------
    for (int i = tid; i < TILE_B * ACC_STRIDE; i += THREADS) acc[i] = 0.0f;
    __syncthreads();

    // ---- Phase 2: scatter-add (memory-bound streaming of from_prev) ----
    for (int e = tid; e < EDGE_N; e += THREADS) {
        const int v = idx[e];
        if (e + THREADS < EDGE_N)   // wave-uniform condition
            __builtin_prefetch(&from_prev[(size_t)b0 * EDGE_N + e + THREADS], 0, 1);
#pragma unroll
        for (int r = 0; r < TILE_B; ++r) {
            // lanes hold consecutive e for fixed r -> coalesced 128B lines
            atomicAdd(&acc[r * ACC_STRIDE + v],
                      from_prev[(size_t)(b0 + r) * EDGE_N + e]);
        }
    }
    __syncthreads();

    // ---- Phase 3: WMMA merge: out = acc * I + from_input (exact fp32) --
    const int lane = tid & 31;
    const int wave = tid >> 5;        // 0..7
    const int nl   = lane & 15;       // N (column) within tile for B/C/D
    const int half = lane >> 4;       // 0 or 1
    const int kb   = half * 2;        // A/B: K base per lane half (K=0,1 | K=2,3)
    const int m    = lane & 15;       // A: M row per lane

    for (int t = wave; t < V_N / 16; t += THREADS / 32) {
        const int n0 = t * 16;
        v8f c;
        // C/D layout: lanes 0-15 -> M = j,   N = lane
        //             lanes16-31 -> M = j+8, N = lane-16
#pragma unroll
        for (int j = 0; j < 8; ++j)
            c[j] = from_input[(size_t)(b0 + half * 8 + j) * V_N + n0 + nl];

#pragma unroll
        for (int j = 0; j < 4; ++j) {
            // A (16x4 f32): lanes 0-15 hold M=0..15 {K=0,K=1}; lanes 16-31 {K=2,K=3}
            v2f a, bb;
            a.x = acc[m * ACC_STRIDE + n0 + 4 * j + kb];
            a.y = acc[m * ACC_STRIDE + n0 + 4 * j + kb + 1];
            // B (4x16): identity slice selecting columns 4j..4j+3
            bb.x = (nl == 4 * j + kb)     ? 1.0f : 0.0f;
            bb.y = (nl == 4 * j + kb + 1) ? 1.0f : 0.0f;
            c = __builtin_amdgcn_wmma_f32_16x16x4_f32(
                    false, a, false, bb, (short)0, c, false, false);
        }

#pragma unroll
        for (int j = 0; j < 8; ++j)
            out[(size_t)(b0 + half * 8 + j) * V_N + n0 + nl] = c[j];
    }
}

extern "C" void kernel_launch(void* const* d_in, const int* in_sizes, int n_in,
                              void* d_out, int out_size, void* d_ws, size_t ws_size,
                              hipStream_t stream) {
    (void)in_sizes; (void)n_in; (void)out_size; (void)ws_size;
    const float* from_input = (const float*)d_in[0];   // [B, V_N]
    const float* from_prev  = (const float*)d_in[1];   // [B, EDGE_N]
    const float* mask       = (const float*)d_in[2];   // [EDGE_N, V_N]
    float* out = (float*)d_out;                        // [B, V_N]
    int*   idx = (int*)d_ws;                           // EDGE_N ints of scratch

    build_edge_index<<<(EDGE_N * V_N) / THREADS, THREADS, 0, stream>>>(mask, idx);

    const size_t shmem = (size_t)TILE_B * ACC_STRIDE * sizeof(float);  // 64 KB
    ldpc_vnode_update<<<B_TOTAL / TILE_B, THREADS, shmem, stream>>>(
        from_input, from_prev, idx, out);
}